// CompleteW2MLSupConLoss_18906446037662
// MI455X (gfx1250) — compile-verified
//
#include <hip/hip_runtime.h>
#include <hip/hip_bf16.h>

#define B_ROWS    8192
#define D_DIM     256
#define N_CLASSES 1024

constexpr float INV_T      = 1.0f / 0.07f;
constexpr float NEG_LOSS_W = 0.3f;

typedef __attribute__((ext_vector_type(2))) float v2f;
typedef __attribute__((ext_vector_type(8))) float v8f;

// ---------------------------------------------------------------- init: zero hist + 2 accumulators
__global__ void w2ml_init(float* hist_acc) {
    int i = blockIdx.x * blockDim.x + threadIdx.x;
    if (i < N_CLASSES + 2) hist_acc[i] = 0.0f;
}

// ---------------------------------------------------------------- label histogram (counts as f32, exact up to 2^24)
__global__ void w2ml_hist(const int* __restrict__ labels, float* __restrict__ hist) {
    int i = blockIdx.x * blockDim.x + threadIdx.x;
    if (i < B_ROWS) atomicAdd(&hist[labels[i]], 1.0f);
}

// ---------------------------------------------------------------- row L2-normalize: one wave32 per row
__global__ void w2ml_normalize(const float* __restrict__ feat, float* __restrict__ anchor) {
    int wave = threadIdx.x >> 5;
    int lane = threadIdx.x & 31;
    int row  = blockIdx.x * 8 + wave;               // blockDim = 256 -> 8 rows per block
    const float* src = feat + (size_t)row * D_DIM + lane * 8;
    float4 a = *(const float4*)(src);
    float4 b = *(const float4*)(src + 4);
    float ss = a.x*a.x + a.y*a.y + a.z*a.z + a.w*a.w
             + b.x*b.x + b.y*b.y + b.z*b.z + b.w*b.w;
    #pragma unroll
    for (int m = 1; m < 32; m <<= 1) ss += __shfl_xor(ss, m, 32);
    float s = 1.0f / sqrtf(ss);
    float* dst = anchor + (size_t)row * D_DIM + lane * 8;
    a.x *= s; a.y *= s; a.z *= s; a.w *= s;
    b.x *= s; b.y *= s; b.z *= s; b.w *= s;
    *(float4*)(dst)     = a;
    *(float4*)(dst + 4) = b;
}

// ---------------------------------------------------------------- fused sim-GEMM (f32 WMMA) + row reductions
// grid = 512 blocks (16 rows each), block = 256 threads (8 waves).
// Each wave handles column tiles jt = wave, wave+8, ... (64 tiles), K=256 via 64 x v_wmma_f32_16x16x4_f32.
__launch_bounds__(256, 1)
__global__ void w2ml_main(const float* __restrict__ anchor,
                          const int*   __restrict__ labels,
                          const float* __restrict__ hist,
                          float*       __restrict__ acc /* [0]=pos_sum, [1]=neg_sum */) {
    __shared__ float sE[16], sP[16], sW[16], sN[16];
    const int tid  = threadIdx.x;
    const int wave = tid >> 5;
    const int lane = tid & 31;
    const int lh   = lane & 15;      // position within half-wave
    const int hi   = lane >> 4;      // which half-wave

    if (tid < 16) { sE[tid] = 0.0f; sP[tid] = 0.0f; sW[tid] = 0.0f; sN[tid] = 0.0f; }
    __syncthreads();

    const int rowBase = blockIdx.x * 16;

    // Preload A fragments for the whole K=256 (64 steps of K=4): 64 x v2f = 128 VGPRs,
    // reused by every column tile this wave processes.
    // f32 16x4 A layout: lanes 0-15 hold M=lane (K = 0,1 in the 2 VGPRs), lanes 16-31 hold K = 2,3.
    const float* arow = anchor + (size_t)(rowBase + lh) * D_DIM + hi * 2;
    v2f aF[64];
    #pragma unroll
    for (int k = 0; k < 64; ++k) aF[k] = *(const v2f*)(arow + k * 4);

    int rowLab[8];
    #pragma unroll
    for (int r = 0; r < 8; ++r) rowLab[r] = labels[rowBase + r + 8 * hi];

    float accE[8], accP[8], accW[8], accN[8];
    #pragma unroll
    for (int r = 0; r < 8; ++r) { accE[r] = 0.f; accP[r] = 0.f; accW[r] = 0.f; accN[r] = 0.f; }

    for (int jt = wave; jt < B_ROWS / 16; jt += 8) {   // uniform per wave -> EXEC all-1s around WMMA
        const int colBase = jt * 16;
        // B (4x16, KxN) fragment for sim = N*N^T is the A-layout of the column rows.
        const float* brow = anchor + (size_t)(colBase + lh) * D_DIM + hi * 2;

        v8f c = {};
        #pragma unroll
        for (int k = 0; k < 64; ++k) {
            v2f bF = *(const v2f*)(brow + k * 4);
            c = __builtin_amdgcn_wmma_f32_16x16x4_f32(
                    /*neg_a=*/false, aF[k], /*neg_b=*/false, bF,
                    /*c_mod=*/(short)0, c, /*reuse_a=*/false, /*reuse_b=*/false);
        }

        const int colLab = labels[colBase + lh];
        #pragma unroll
        for (int r = 0; r < 8; ++r) {
            const float sim  = c[r];
            const int   m    = r + 8 * hi;           // C layout: vgpr r, halves -> rows r / r+8
            const bool  diag = (rowBase + m) == (colBase + lh);
            const bool  same = (rowLab[r] == colLab);
            const float l    = sim * INV_T;
            const float e    = __expf(l);
            accE[r] += diag ? 0.0f : e;
            // positive (same class, off-diagonal): w = sim<0.5 ? 1.5-sim : 1
            const float wp = (sim < 0.5f) ? (1.5f - sim) : 1.0f;
            const float pg = (same && !diag) ? 1.0f : 0.0f;
            accP[r] += pg * wp * l;
            accW[r] += pg * wp;
            // negative (different class): w = sim>0.3 ? 1+(sim-0.3)/0.7 : 1
            const float wn = (sim > 0.3f) ? (1.0f + (sim - 0.3f) * (1.0f / 0.7f)) : 1.0f;
            accN[r] += same ? 0.0f : wn * e;
        }
    }

    // Reduce across the 16 lanes of each half-wave (xor masks <16 stay within the half).
    #pragma unroll
    for (int r = 0; r < 8; ++r) {
        for (int m = 1; m < 16; m <<= 1) {
            accE[r] += __shfl_xor(accE[r], m, 32);
            accP[r] += __shfl_xor(accP[r], m, 32);
            accW[r] += __shfl_xor(accW[r], m, 32);
            accN[r] += __shfl_xor(accN[r], m, 32);
        }
    }
    // One representative lane per half-wave merges into LDS (ds_add_f32).
    if (lh == 0) {
        #pragma unroll
        for (int r = 0; r < 8; ++r) {
            const int rt = r + 8 * hi;
            atomicAdd(&sE[rt], accE[r]);
            atomicAdd(&sP[rt], accP[r]);
            atomicAdd(&sW[rt], accW[r]);
            atomicAdd(&sN[rt], accN[r]);
        }
    }
    __syncthreads();

    // Finalize the block's 16 rows and atomically add into the two global scalars.
    if (tid < 16) {
        const int   row = rowBase + tid;
        const float E   = sE[tid];
        const float P   = sP[tid];
        const float W   = sW[tid];
        const float Nn  = sN[tid];
        const float cp  = hist[labels[row]] - 1.0f;           // # same-class, self excluded
        const float cn  = (float)(B_ROWS - 1) - cp;           // # negatives
        const float cpc = fmaxf(cp, 1.0f);
        const float cnc = fmaxf(cn, 1.0f);
        float posL = -(P - W * __logf(E)) / cpc;
        float negL = (Nn / E) / cnc;
        for (int m = 1; m < 16; m <<= 1) {
            posL += __shfl_xor(posL, m, 32);
            negL += __shfl_xor(negL, m, 32);
        }
        if (tid == 0) {
            atomicAdd(&acc[0], posL);
            atomicAdd(&acc[1], negL);
        }
    }
}

// ---------------------------------------------------------------- final scalar
__global__ void w2ml_final(const float* __restrict__ acc, float* __restrict__ out) {
    if (threadIdx.x == 0 && blockIdx.x == 0) {
        out[0] = acc[0] * (1.0f / B_ROWS) + NEG_LOSS_W * (acc[1] * (1.0f / B_ROWS));
    }
}

// ----------------------------------------------------------------
extern "C" void kernel_launch(void* const* d_in, const int* in_sizes, int n_in,
                              void* d_out, int out_size, void* d_ws, size_t ws_size,
                              hipStream_t stream) {
    (void)in_sizes; (void)n_in; (void)out_size; (void)ws_size;
    const float* feat   = (const float*)d_in[0];
    const int*   labels = (const int*)d_in[1];
    float*       out    = (float*)d_out;

    // workspace layout: anchor[8192*256] | hist[1024] | acc[2]
    float* anchor = (float*)d_ws;
    float* hist   = anchor + (size_t)B_ROWS * D_DIM;
    float* acc    = hist + N_CLASSES;

    w2ml_init<<<(N_CLASSES + 2 + 255) / 256, 256, 0, stream>>>(hist);
    w2ml_hist<<<B_ROWS / 256, 256, 0, stream>>>(labels, hist);
    w2ml_normalize<<<B_ROWS / 8, 256, 0, stream>>>(feat, anchor);
    w2ml_main<<<B_ROWS / 16, 256, 0, stream>>>(anchor, labels, hist, acc);
    w2ml_final<<<1, 64, 0, stream>>>(acc, out);
}